// HIMOSAv1_37117107372448
// MI455X (gfx1250) — compile-verified
//
#include <hip/hip_runtime.h>
#include <math.h>

#define B_    8
#define Hn    256
#define Wn    256
#define C_    180
#define HEADS 6
#define HDIM  30
#define L_    256
#define K_    64
#define BW    2048   // B_ * (Hn/16) * (Wn/16)

typedef float v2f __attribute__((ext_vector_type(2)));
typedef float v8f __attribute__((ext_vector_type(8)));

static __device__ __forceinline__ v8f wmma4(v2f a, v2f b, v8f c) {
  // D = A(16x4 f32) * B(4x16 f32) + C(16x16 f32)
  return __builtin_amdgcn_wmma_f32_16x16x4_f32(false, a, false, b, (short)0, c,
                                               false, false);
}

// ---------------------------------------------------------------------------
// Kernel 1: router logits + sigmoid + rank-based top-(k-1) (plus token 0)
// One block per window, 256 threads (one per token).
// ---------------------------------------------------------------------------
__global__ __launch_bounds__(256) void router_topk(
    const float* __restrict__ x, const float* __restrict__ Wr,
    int* __restrict__ idx_out, float* __restrict__ gate_out) {
  __shared__ float wr[HEADS * C_];
  __shared__ float slog[HEADS * L_];
  __shared__ int scnt;
  const int tid = threadIdx.x;
  const int win = blockIdx.x;
  for (int i = tid; i < HEADS * C_; i += 256) wr[i] = Wr[i];
  __syncthreads();

  const int b = win >> 8, wh = (win >> 4) & 15, wwi = win & 15;
  const int r = tid >> 4, cc = tid & 15;
  const long pix = ((long)(b * Hn + wh * 16 + r) * Wn + (wwi * 16 + cc));
  const float4* xr = (const float4*)(x + pix * C_);  // 720B rows, 16B aligned

  float z[HEADS];
#pragma unroll
  for (int e = 0; e < HEADS; ++e) z[e] = 0.f;
  for (int c4 = 0; c4 < C_ / 4; ++c4) {
    float4 xv = xr[c4];
#pragma unroll
    for (int e = 0; e < HEADS; ++e) {
      const float* w = &wr[e * C_ + c4 * 4];
      z[e] += xv.x * w[0] + xv.y * w[1] + xv.z * w[2] + xv.w * w[3];
    }
  }
#pragma unroll
  for (int e = 0; e < HEADS; ++e)
    slog[e * L_ + tid] = 1.0f / (1.0f + __expf(-z[e]));
  __syncthreads();

  for (int e = 0; e < HEADS; ++e) {
    if (tid == 0) scnt = 0;
    __syncthreads();
    const float my = slog[e * L_ + tid];
    int rank = 0;
    if (tid > 0) {  // rank among tokens 1..255 with total-order tie-break
      for (int j = 1; j < L_; ++j) {
        float vj = slog[e * L_ + j];
        if (vj > my || (vj == my && j < tid)) ++rank;
      }
    }
    const bool sel = (tid == 0) || (rank < K_ - 1);
    if (sel) {
      int slot = atomicAdd(&scnt, 1);  // order irrelevant to final output
      idx_out[((long)win * HEADS + e) * K_ + slot] = tid;
      gate_out[((long)win * HEADS + e) * K_ + slot] = my;
    }
    __syncthreads();
  }
}

// ---------------------------------------------------------------------------
// Kernel 2: per-window gather + QKV + attention + Wo + LDS scatter-add + y
// One block per window, 256 threads = 8 wave32s. ~140KB LDS (CDNA5 WGP: 320KB).
// ---------------------------------------------------------------------------
__global__ __launch_bounds__(256) void moe_attn(
    const float* __restrict__ x, const float* __restrict__ Wqkv,
    const float* __restrict__ Wo, const int* __restrict__ idx_in,
    const float* __restrict__ gate_in, float* __restrict__ y) {
  __shared__ float smem[34304];      // 12288 (O_all) + 22016 (scratch)
  __shared__ int sidx[HEADS * K_];
  __shared__ float sgate[HEADS * K_];
  float* const O_all = smem;               // [6][64][32] gated attn outputs
  float* const xg = smem + 12288;          // [64][184]   gathered tokens
  float* const qm = smem + 12288 + 11776;  // [64][32]    Q (cols 30,31 = 0)
  float* const km = qm + 2048;             // [64][32]    K
  float* const vm = km + 2048;             // [64][32]    V
  float* const sm = smem + 12288 + 17920;  // [64][64]    scores / probs
  float* const ych = smem + 12288;         // [256][64]   phase-2 y chunk

  const int tid = threadIdx.x;
  const int wave = tid >> 5;
  const int lane = tid & 31;
  const int l16 = lane & 15;
  const int half = lane >> 4;
  const int win = blockIdx.x;
  const int b = win >> 8, wh = (win >> 4) & 15, wwi = win & 15;

  if (tid < HEADS * K_) {
    sidx[tid] = idx_in[(long)win * HEADS * K_ + tid];
    sgate[tid] = gate_in[(long)win * HEADS * K_ + tid];
  }
  __syncthreads();
  const float scale = 0.18257418583505536f;  // 1/sqrt(30)

  // ---------------- phase 1: per-head attention, keep gated O in LDS -------
  for (int e = 0; e < HEADS; ++e) {
    __syncthreads();
    __builtin_prefetch(Wqkv + (long)e * C_ * 90 + tid * 64, 0, 3);
    for (int i = tid; i < 3 * 64 * 32; i += 256) qm[i] = 0.f;  // q/k/v zero
    for (int i = tid; i < K_ * C_; i += 256) {                 // gather
      int m = i / C_, c = i - m * C_;
      int tok = sidx[e * K_ + m];
      int rr = tok >> 4, cc2 = tok & 15;
      long pix = ((long)(b * Hn + wh * 16 + rr) * Wn + (wwi * 16 + cc2));
      xg[m * 184 + c] = x[pix * C_ + c];
    }
    __syncthreads();
    // QKV = Xg(64x180) @ Wqkv[e](180x90), N padded to 96 -> 24 tiles
    for (int t = wave * 3; t < wave * 3 + 3; ++t) {
      const int m0 = (t & 3) * 16;
      const int n = (t >> 2) * 16 + l16;
      v8f acc = {};
      for (int k0 = 0; k0 < C_; k0 += 4) {
        const int ka = k0 + 2 * half;
        v2f a;  a.x = xg[(m0 + l16) * 184 + ka];
                a.y = xg[(m0 + l16) * 184 + ka + 1];
        v2f bb;
        bb.x = (n < 90) ? Wqkv[((long)e * C_ + ka) * 90 + n] : 0.f;
        bb.y = (n < 90) ? Wqkv[((long)e * C_ + ka + 1) * 90 + n] : 0.f;
        acc = wmma4(a, bb, acc);
      }
#pragma unroll
      for (int rr = 0; rr < 8; ++rr) {
        int m = m0 + rr + 8 * half;
        if (n < 30)      qm[m * 32 + n] = acc[rr];
        else if (n < 60) km[m * 32 + (n - 30)] = acc[rr];
        else if (n < 90) vm[m * 32 + (n - 60)] = acc[rr];
      }
    }
    __syncthreads();
    // scores = Q @ K^T, K-dim padded 30->32 -> 16 tiles
    for (int t = wave * 2; t < wave * 2 + 2; ++t) {
      const int m0 = (t & 3) * 16;
      const int n0 = (t >> 2) * 16;
      v8f acc = {};
      for (int k0 = 0; k0 < 32; k0 += 4) {
        const int ka = k0 + 2 * half;
        v2f a;  a.x = qm[(m0 + l16) * 32 + ka]; a.y = qm[(m0 + l16) * 32 + ka + 1];
        v2f bb; bb.x = km[(n0 + l16) * 32 + ka]; bb.y = km[(n0 + l16) * 32 + ka + 1];
        acc = wmma4(a, bb, acc);
      }
#pragma unroll
      for (int rr = 0; rr < 8; ++rr)
        sm[(m0 + rr + 8 * half) * 64 + n0 + l16] = acc[rr] * scale;
    }
    __syncthreads();
    if (tid < K_) {  // softmax, one thread per row
      float mx = -1e30f;
      for (int j = 0; j < K_; ++j) mx = fmaxf(mx, sm[tid * 64 + j]);
      float sum = 0.f;
      for (int j = 0; j < K_; ++j) {
        float ev = __expf(sm[tid * 64 + j] - mx);
        sm[tid * 64 + j] = ev; sum += ev;
      }
      float inv = 1.f / sum;
      for (int j = 0; j < K_; ++j) sm[tid * 64 + j] *= inv;
    }
    __syncthreads();
    // O = P(64x64) @ V(64x32padded), gated -> 8 tiles, one per wave
    {
      const int m0 = (wave & 3) * 16;
      const int n0 = (wave >> 2) * 16;
      v8f acc = {};
      for (int k0 = 0; k0 < 64; k0 += 4) {
        const int ka = k0 + 2 * half;
        v2f a;  a.x = sm[(m0 + l16) * 64 + ka]; a.y = sm[(m0 + l16) * 64 + ka + 1];
        v2f bb; bb.x = vm[ka * 32 + n0 + l16];  bb.y = vm[(ka + 1) * 32 + n0 + l16];
        acc = wmma4(a, bb, acc);
      }
#pragma unroll
      for (int rr = 0; rr < 8; ++rr) {
        int m = m0 + rr + 8 * half;
        O_all[(e * K_ + m) * 32 + n0 + l16] = acc[rr] * sgate[e * K_ + m];
      }
    }
  }
  __syncthreads();

  // ---------------- phase 2: pre = O @ Wo[e], scatter-add in LDS, write y --
  for (int cc0 = 0; cc0 < C_; cc0 += 64) {
    for (int i = tid; i < 256 * 64; i += 256) ych[i] = 0.f;
    __syncthreads();
    for (int e = 0; e < HEADS; ++e) {
      for (int t = wave * 2; t < wave * 2 + 2; ++t) {
        const int m0 = (t & 3) * 16;
        const int n0 = (t >> 2) * 16;
        const int n = cc0 + n0 + l16;
        v8f acc = {};
        for (int k0 = 0; k0 < 32; k0 += 4) {
          const int ka = k0 + 2 * half;
          v2f a;  a.x = O_all[(e * K_ + m0 + l16) * 32 + ka];
                  a.y = O_all[(e * K_ + m0 + l16) * 32 + ka + 1];
          v2f bb;
          bb.x = (ka < HDIM && n < C_) ? Wo[((long)e * HDIM + ka) * C_ + n] : 0.f;
          bb.y = (ka + 1 < HDIM && n < C_) ? Wo[((long)e * HDIM + ka + 1) * C_ + n] : 0.f;
          acc = wmma4(a, bb, acc);
        }
#pragma unroll
        for (int rr = 0; rr < 8; ++rr) {  // idx unique within head: no race
          int m = m0 + rr + 8 * half;
          ych[sidx[e * K_ + m] * 64 + n0 + l16] += acc[rr];
        }
      }
      __syncthreads();  // heads overlap in token indices: serialize
    }
    for (int i = tid; i < 256 * 64; i += 256) {  // fused window_reverse store
      int tok = i >> 6, nl = i & 63;
      int c = cc0 + nl;
      if (c < C_) {
        int rr = tok >> 4, cc2 = tok & 15;
        long pix = ((long)(b * Hn + wh * 16 + rr) * Wn + (wwi * 16 + cc2));
        y[pix * C_ + c] = ych[i];
      }
    }
    __syncthreads();
  }
}

// ---------------------------------------------------------------------------
// Kernel 3: out = y @ Wp^T + bp  (524288 x 180 x 180)
// 128 rows per block staged coalesced into LDS; 8 waves x 12 n-tiles.
// ---------------------------------------------------------------------------
__global__ __launch_bounds__(256) void proj(
    const float* __restrict__ y, const float* __restrict__ Wp,
    const float* __restrict__ bp, float* __restrict__ out) {
  __shared__ float yt[128 * 184];
  const int tid = threadIdx.x;
  const int wave = tid >> 5;
  const int lane = tid & 31;
  const int l16 = lane & 15;
  const int half = lane >> 4;
  const long base = (long)blockIdx.x * 128;
  for (int i = tid; i < 128 * C_; i += 256) {
    int rr = i / C_, c = i - rr * C_;
    yt[rr * 184 + c] = y[(base + rr) * C_ + c];
  }
  __syncthreads();
  const int m0 = wave * 16;
  for (int nt = 0; nt < 12; ++nt) {
    const int n = nt * 16 + l16;
    v8f acc = {};
    for (int k0 = 0; k0 < C_; k0 += 4) {
      const int ka = k0 + 2 * half;
      v2f a;  a.x = yt[(m0 + l16) * 184 + ka]; a.y = yt[(m0 + l16) * 184 + ka + 1];
      v2f bb;
      bb.x = (n < C_) ? Wp[(long)n * C_ + ka] : 0.f;       // B[k][n] = Wp[n][k]
      bb.y = (n < C_) ? Wp[(long)n * C_ + ka + 1] : 0.f;
      acc = wmma4(a, bb, acc);
    }
    if (n < C_) {
      const float bias = bp[n];
#pragma unroll
      for (int rr = 0; rr < 8; ++rr) {
        long row = base + m0 + rr + 8 * half;
        out[row * C_ + n] = acc[rr] + bias;
      }
    }
  }
}

// ---------------------------------------------------------------------------
extern "C" void kernel_launch(void* const* d_in, const int* in_sizes, int n_in,
                              void* d_out, int out_size, void* d_ws,
                              size_t ws_size, hipStream_t stream) {
  const float* x    = (const float*)d_in[0];
  const float* Wr   = (const float*)d_in[1];
  const float* Wqkv = (const float*)d_in[2];
  const float* Wo   = (const float*)d_in[3];
  const float* Wp   = (const float*)d_in[4];
  const float* bp   = (const float*)d_in[5];
  float* out = (float*)d_out;

  const size_t yElems = (size_t)B_ * Hn * Wn * C_;           // 94,371,840
  float* yb   = (float*)d_ws;
  int*   idxb = (int*)((char*)d_ws + yElems * sizeof(float));
  float* gateb = (float*)((char*)idxb + (size_t)BW * HEADS * K_ * sizeof(int));

  router_topk<<<BW, 256, 0, stream>>>(x, Wr, idxb, gateb);
  moe_attn<<<BW, 256, 0, stream>>>(x, Wqkv, Wo, idxb, gateb, yb);
  proj<<<(B_ * Hn * Wn) / 128, 256, 0, stream>>>(yb, Wp, bp, out);
}